// Generator_13400297963929
// MI455X (gfx1250) — compile-verified
//
#include <hip/hip_runtime.h>
#include <math.h>

// ---------------------------------------------------------------------------
// 22-qubit statevector circuit on MI455X (gfx1250).
//
// Design (see analysis): CNOT rings are index permutations fused into gathers;
// rotation layers are Kronecker products applied as six 16x16 complex GEMM
// passes per layer using V_WMMA_F32_16X16X4_F32. 32MB state stays L2-resident.
// ---------------------------------------------------------------------------

typedef __attribute__((ext_vector_type(2))) float v2f;
typedef __attribute__((ext_vector_type(8))) float v8f;

#define NQ 22
#define NSTATE (1u << NQ)

__device__ __forceinline__ float2 cmul(float2 a, float2 b) {
  return make_float2(a.x * b.x - a.y * b.y, a.x * b.y + a.y * b.x);
}

// Full CNOT ring (q -> q+1 mod 22, qubit q lives at flat bit 21-q) as a
// read-side index permutation: out[i] = in[perm(i)].
__device__ __forceinline__ uint32_t ring_perm(uint32_t i) {
  uint32_t r = ((i >> 1) | (i << (NQ - 1))) & (NSTATE - 1u);
  return (i ^ r) ^ ((i & 1u) << 20);
}

// Flat amplitude index with 4 "group" bits m inserted at bit position b,
// remaining 18 bits taken from o.
__device__ __forceinline__ uint32_t amp_index(uint32_t o, uint32_t m, int b) {
  uint32_t low = o & ((1u << b) - 1u);
  return low | (m << b) | ((o >> b) << (b + 4));
}

// ---------------------------------------------------------------------------
// Prep: fuse RZ-RX-RZ per qubit into 2x2 gates; build the six 16x16 fused
// group matrices (stored transposed: MT[k][m] = M[m][k]) for layers 1 and 2.
// ---------------------------------------------------------------------------
__global__ void prep_kernel(const float* __restrict__ p, float2* __restrict__ gates,
                            float* __restrict__ MTre, float* __restrict__ MTim) {
  __shared__ float2 g[3 * 22 * 4];
  int t = threadIdx.x;
  if (t < 66) {
    int l = t / 22, q = t % 22;
    float a  = 0.5f * p[l * 66 + 3 * q + 0];
    float bb = 0.5f * p[l * 66 + 3 * q + 1];
    float c  = 0.5f * p[l * 66 + 3 * q + 2];
    float cb = cosf(bb), sb = sinf(bb);
    // m00 = e^{-i(a+c)} cos b ; m01 = -i e^{-i(c-a)} sin b
    // m10 = -i e^{-i(a-c)} sin b ; m11 = e^{+i(a+c)} cos b
    float2 m00 = make_float2(cosf(a + c) * cb, -sinf(a + c) * cb);
    float2 m01 = make_float2(-sb * sinf(c - a), -sb * cosf(c - a));
    float2 m10 = make_float2(-sb * sinf(a - c), -sb * cosf(a - c));
    float2 m11 = make_float2(cosf(a + c) * cb, sinf(a + c) * cb);
    int base = t * 4;
    g[base + 0] = m00; g[base + 1] = m01; g[base + 2] = m10; g[base + 3] = m11;
    gates[base + 0] = m00; gates[base + 1] = m01;
    gates[base + 2] = m10; gates[base + 3] = m11;
  }
  __syncthreads();
  // 2 layers x 6 groups x 16x16 entries
  for (int e = t; e < 2 * 6 * 256; e += blockDim.x) {
    int l   = 1 + e / 1536;
    int rem = e % 1536;
    int grp = rem / 256;
    int km  = rem % 256;
    int k = km >> 4, m = km & 15;
    int bsh = (grp == 5) ? 18 : grp * 4;
    float2 acc = make_float2(1.f, 0.f);
#pragma unroll
    for (int j = 0; j < 4; ++j) {
      int beta = bsh + j;           // flat bit position; qubit = 21-beta
      int mb = (m >> j) & 1, kb = (k >> j) & 1;
      float2 ge;
      if (grp == 5 && j < 2) {      // identity padding on bits 18,19
        ge = make_float2((mb == kb) ? 1.f : 0.f, 0.f);
      } else {
        ge = g[(l * 22 + (21 - beta)) * 4 + mb * 2 + kb];
      }
      acc = cmul(acc, ge);
    }
    MTre[e] = acc.x;  // index e == [(l-1)][grp][k][m]
    MTim[e] = acc.y;
  }
}

// ---------------------------------------------------------------------------
// Init: S[i] = (ring o layer0)|0..0> = prod_q G0[q][bit_q(perm(i)), 0].
// ---------------------------------------------------------------------------
__global__ void init_kernel(const float2* __restrict__ gates, float2* __restrict__ S) {
  __shared__ float2 gc[44];  // column-0 entries: [q][row]
  int t = threadIdx.x;
  if (t < 44) gc[t] = gates[(t >> 1) * 4 + (t & 1) * 2];
  __syncthreads();
  uint32_t i = blockIdx.x * blockDim.x + threadIdx.x;
  uint32_t j = ring_perm(i);
  float2 acc = make_float2(1.f, 0.f);
#pragma unroll
  for (int q = 0; q < NQ; ++q) {
    uint32_t bit = (j >> (21 - q)) & 1u;
    acc = cmul(acc, gc[q * 2 + bit]);
  }
  S[i] = acc;
}

// ---------------------------------------------------------------------------
// One fused-group pass: out(o, m) = sum_k MT[k][m] * in(o, k) for all 2^18 o.
// Each wave owns a 16(o) x 16(k->m) tile; complex GEMM via 16 f32 WMMAs.
// In-place safe (tiles disjoint per wave). do_perm fuses the CNOT-ring gather.
// ---------------------------------------------------------------------------
__global__ void group_pass(const float2* src, float2* dst,
                           const float* __restrict__ MTre,
                           const float* __restrict__ MTim,
                           int b, int do_perm) {
  int lane = threadIdx.x & 31;
  uint32_t wave = (blockIdx.x * blockDim.x + threadIdx.x) >> 5;
  uint32_t o0 = wave << 4;
  int half = lane >> 4;   // lanes 16-31 hold K+2 (A) / rows M+8 (D)
  int lr = lane & 15;

  v8f Dr = {};
  v8f Di = {};

#pragma unroll
  for (int c = 0; c < 4; ++c) {            // K chunks of 4
    int k0 = c * 4 + 2 * half;             // this lane-half's K base
    uint32_t ia0 = amp_index(o0 + (uint32_t)lr, (uint32_t)k0, b);
    uint32_t ia1 = amp_index(o0 + (uint32_t)lr, (uint32_t)(k0 + 1), b);
    if (do_perm) { ia0 = ring_perm(ia0); ia1 = ring_perm(ia1); }
    float2 s0 = src[ia0];
    float2 s1 = src[ia1];
    v2f Ar  = { s0.x, s1.x };
    v2f Ai  = { s0.y, s1.y };
    v2f AiN = { -s0.y, -s1.y };
    v2f Br  = { MTre[k0 * 16 + lr], MTre[(k0 + 1) * 16 + lr] };
    v2f Bi  = { MTim[k0 * 16 + lr], MTim[(k0 + 1) * 16 + lr] };
    // D = A x B + C ; complex: Dr += Ar*Br - Ai*Bi ; Di += Ar*Bi + Ai*Br
    Dr = __builtin_amdgcn_wmma_f32_16x16x4_f32(false, Ar,  false, Br, (short)0, Dr, false, false);
    Dr = __builtin_amdgcn_wmma_f32_16x16x4_f32(false, AiN, false, Bi, (short)0, Dr, false, false);
    Di = __builtin_amdgcn_wmma_f32_16x16x4_f32(false, Ar,  false, Bi, (short)0, Di, false, false);
    Di = __builtin_amdgcn_wmma_f32_16x16x4_f32(false, Ai,  false, Br, (short)0, Di, false, false);
  }

#pragma unroll
  for (int r = 0; r < 8; ++r) {            // D VGPR r: rows r (lanes<16) / r+8
    uint32_t orow = o0 + (uint32_t)(r + 8 * half);
    uint32_t idx = amp_index(orow, (uint32_t)lr, b);
    dst[idx] = make_float2(Dr[r], Di[r]);
  }
}

__global__ void probs_kernel(const float2* __restrict__ S, float* __restrict__ out) {
  uint32_t i = blockIdx.x * blockDim.x + threadIdx.x;
  float2 a = S[i];
  out[i] = a.x * a.x + a.y * a.y;
}

// ---------------------------------------------------------------------------
extern "C" void kernel_launch(void* const* d_in, const int* in_sizes, int n_in,
                              void* d_out, int out_size, void* d_ws, size_t ws_size,
                              hipStream_t stream) {
  const float* params = (const float*)d_in[0];  // (3, 66) float32

  char* base = (char*)d_ws;
  float2* S = (float2*)base;                                   // 32 MB
  float2* T = S + (size_t)NSTATE;                              // 32 MB
  float2* gates = T + (size_t)NSTATE;                          // 3*22*4 float2
  float* MTre = (float*)(gates + 3 * 22 * 4);                  // 2*6*256 f32
  float* MTim = MTre + 2 * 6 * 256;

  prep_kernel<<<1, 256, 0, stream>>>(params, gates, MTre, MTim);
  init_kernel<<<NSTATE / 256, 256, 0, stream>>>(gates, S);

  static const int bshift[6] = {0, 4, 8, 12, 16, 18};
  const int blocks = (1 << 14) * 32 / 256;  // 2^14 waves per pass -> 2048 blocks

  // Layer 1 rotations: six in-place group passes on S.
  for (int g = 0; g < 6; ++g)
    group_pass<<<blocks, 256, 0, stream>>>(S, S, MTre + g * 256, MTim + g * 256,
                                           bshift[g], 0);
  // Ring permutation fused into first layer-2 pass (S -> T), rest in-place.
  group_pass<<<blocks, 256, 0, stream>>>(S, T, MTre + (6 + 0) * 256,
                                         MTim + (6 + 0) * 256, bshift[0], 1);
  for (int g = 1; g < 6; ++g)
    group_pass<<<blocks, 256, 0, stream>>>(T, T, MTre + (6 + g) * 256,
                                           MTim + (6 + g) * 256, bshift[g], 0);

  probs_kernel<<<NSTATE / 256, 256, 0, stream>>>(T, (float*)d_out);
}